// GAT_22016002359723
// MI455X (gfx1250) — compile-verified
//
#include <hip/hip_runtime.h>
#include <hip/hip_bf16.h>

typedef __attribute__((ext_vector_type(2))) float    v2f;
typedef __attribute__((ext_vector_type(8))) float    v8f;
typedef __attribute__((ext_vector_type(4))) unsigned tdm_v4u;
typedef __attribute__((ext_vector_type(4))) int      tdm_v4i;
typedef __attribute__((ext_vector_type(8))) int      tdm_v8i;

#define NEG_SLOPE 0.2f
#define SM_EPS    1e-16f
#define KCHUNK    32

// ---------------- order-preserving float<->uint for atomicMax ----------------
__device__ __forceinline__ unsigned enc_ord(float f) {
    unsigned b = __float_as_uint(f);
    return (b & 0x80000000u) ? ~b : (b | 0x80000000u);
}
__device__ __forceinline__ float dec_ord(unsigned u) {
    unsigned b = (u & 0x80000000u) ? (u ^ 0x80000000u) : ~u;
    return __uint_as_float(b);
}

__device__ __forceinline__ void get_edge(const long long* __restrict__ srcp,
                                         const long long* __restrict__ dstp,
                                         int e, int E, int& s, int& d) {
    if (e < E) { s = (int)srcp[e]; d = (int)dstp[e]; }
    else       { s = e - E; d = e - E; }          // self loops appended
}

// ---------------- TDM: DMA a rows x cols f32 panel (global -> LDS) ----------
// D# built per CDNA5 ISA ch.8: group0 = {count/flags, lds_addr, gaddr, type},
// group1 = {data_size, tensor dims, tile dims, dim0 stride}. 2D load, no pad,
// no multicast, no atomic-barrier. Issued by one wave; tracked by TENSORcnt.
__device__ __forceinline__ void tdm_load_2d_f32(const float* gsrc,
                                                unsigned lds_byte_off,
                                                unsigned rows, unsigned cols,
                                                unsigned stride_elems) {
    unsigned long long ga = (unsigned long long)(size_t)gsrc;
    tdm_v4u g0;
    g0[0] = 1u;                                              // count=1, user D#
    g0[1] = lds_byte_off;                                    // lds_addr (bytes)
    g0[2] = (unsigned)ga;                                    // global_addr[31:0]
    g0[3] = (unsigned)((ga >> 32) & 0x01FFFFFFu) | (2u << 30); // addr[56:32]|type=2
    tdm_v8i g1;
    g1[0] = (int)(2u << 16);                                 // data_size=4B, mask=0
    g1[1] = (int)((cols & 0xFFFFu) << 16);                   // tensor_dim0[15:0]
    g1[2] = (int)((cols >> 16) | ((rows & 0xFFFFu) << 16));  // dim0 hi | dim1 lo
    g1[3] = (int)((rows >> 16) | ((cols & 0xFFFFu) << 16));  // dim1 hi | tile_dim0
    g1[4] = (int)(rows & 0xFFFFu);                           // tile_dim1 (tile_dim2=0)
    g1[5] = (int)stride_elems;                               // tensor_dim0_stride lo
    g1[6] = 0;
    g1[7] = 0;
    tdm_v4i z4 = {0, 0, 0, 0};
#if __has_include(<hip/amd_detail/amd_gfx1250_TDM.h>)
    tdm_v8i z8 = {0, 0, 0, 0, 0, 0, 0, 0};
    __builtin_amdgcn_tensor_load_to_lds(g0, g1, z4, z4, z8, 0);   // clang-23 arity
#else
    __builtin_amdgcn_tensor_load_to_lds(g0, g1, z4, z4, 0);       // ROCm 7.2 arity
#endif
}

// ---------------- fp32 WMMA GEMM with TDM-staged B in LDS -------------------
// C[M, TILES_N*16] = A[M,K] @ B[K, TILES_N*16]. One wave owns a 16-row strip
// and all TILES_N column tiles (one A-frag feeds TILES_N WMMAs). B is DMA'd
// to LDS in KCHUNK-row panels by wave 0 (tensor_load_to_lds + s_wait_tensorcnt),
// WMMA B-fragments come from ds_load. Requires M%16==0, K%KCHUNK==0.
template <int TILES_N>
__global__ void __launch_bounds__(256)
gat_gemm_wmma_lds(const float* __restrict__ A, const float* __restrict__ B,
                  float* __restrict__ C, int M, int K) {
    constexpr int N = TILES_N * 16;
    __shared__ float ldsB[KCHUNK * N];                // 32KB (L1) / 8KB (L2)

    const int waveInBlk = threadIdx.x >> 5;
    const int lane      = threadIdx.x & 31;
    const int l15       = lane & 15;
    const int hi        = lane >> 4;                  // 0: K={0,1}, 1: K={2,3}
    const long long gwave = (long long)blockIdx.x * 8 + waveInBlk;
    const bool active   = gwave < (M >> 4);           // wave-uniform predicate
    const int  tm       = (int)(gwave << 4);

    v8f acc[TILES_N] = {};
    const float* arow = A + (size_t)((active ? tm : 0) + l15) * K;

    for (int k0 = 0; k0 < K; k0 += KCHUNK) {
        if (waveInBlk == 0) {                         // single-wave DMA issue
            tdm_load_2d_f32(B + (size_t)k0 * N,
                            (unsigned)(size_t)&ldsB[0],
                            KCHUNK, N, N);
            __builtin_amdgcn_s_wait_tensorcnt(0);     // TENSORcnt is per-wave
        }
        __syncthreads();                              // publish panel to block
        if (active) {
#pragma unroll
            for (int kk = 0; kk < KCHUNK; kk += 4) {
                const int ka = k0 + kk + (hi << 1);
                v2f a;
                a.x = arow[ka];
                a.y = arow[ka + 1];
                const float* bp = &ldsB[(kk + (hi << 1)) * N + l15];
#pragma unroll
                for (int t = 0; t < TILES_N; ++t) {   // A-frag reused TILES_N x
                    v2f b;
                    b.x = bp[t * 16];
                    b.y = bp[t * 16 + N];
                    acc[t] = __builtin_amdgcn_wmma_f32_16x16x4_f32(
                        false, a, false, b, (short)0, acc[t], false, false);
                }
            }
        }
        __syncthreads();                              // panel reuse barrier
    }

    if (active) {
#pragma unroll
        for (int t = 0; t < TILES_N; ++t) {
            float* cp = C + (size_t)(tm + hi * 8) * N + t * 16 + l15;
#pragma unroll
            for (int r = 0; r < 8; ++r) cp[(size_t)r * N] = acc[t][r];
        }
    }
}

// ---------------- per-(node,head) attention scalars --------------------------
__global__ void gat_att_scores(const float* __restrict__ h,
                               const float* __restrict__ att_s,
                               const float* __restrict__ att_d,
                               float* __restrict__ a_src,
                               float* __restrict__ a_dst,
                               int Nn, int H, int C) {
    int i = blockIdx.x * blockDim.x + threadIdx.x;
    if (i >= Nn * H) return;
    int n = i / H, hh = i % H;
    const float* hp = h + (size_t)n * H * C + (size_t)hh * C;
    const float* ws = att_s + hh * C;
    const float* wd = att_d + hh * C;
    float ss = 0.f, sd = 0.f;
    for (int c = 0; c < C; ++c) { float v = hp[c]; ss += v * ws[c]; sd += v * wd[c]; }
    a_src[i] = ss;
    a_dst[i] = sd;
}

// ---------------- edge pass 1: segment max (leaky-relu'd logits) -------------
__global__ void gat_edge_max(const float* __restrict__ a_src,
                             const float* __restrict__ a_dst,
                             unsigned* __restrict__ mu,
                             const long long* __restrict__ srcp,
                             const long long* __restrict__ dstp,
                             int E, int Nn, int H) {
    int i = blockIdx.x * blockDim.x + threadIdx.x;
    int tot = (E + Nn) * H;
    if (i >= tot) return;
    int e = i / H, hh = i - e * H;
    int s, d; get_edge(srcp, dstp, e, E, s, d);
    float v = a_src[s * H + hh] + a_dst[d * H + hh];
    v = v > 0.f ? v : NEG_SLOPE * v;
    atomicMax(&mu[d * H + hh], enc_ord(v));
}

// ---------------- edge pass 2: segment sum of exp(e - m) ---------------------
__global__ void gat_edge_sum(const float* __restrict__ a_src,
                             const float* __restrict__ a_dst,
                             const unsigned* __restrict__ mu,
                             float* __restrict__ ssum,
                             const long long* __restrict__ srcp,
                             const long long* __restrict__ dstp,
                             int E, int Nn, int H) {
    int i = blockIdx.x * blockDim.x + threadIdx.x;
    int tot = (E + Nn) * H;
    if (i >= tot) return;
    int e = i / H, hh = i - e * H;
    int s, d; get_edge(srcp, dstp, e, E, s, d);
    float v = a_src[s * H + hh] + a_dst[d * H + hh];
    v = v > 0.f ? v : NEG_SLOPE * v;
    float w = __expf(v - dec_ord(mu[d * H + hh]));
    atomicAdd(&ssum[d * H + hh], w);
}

// ---------------- edge pass 3: alpha-weighted scatter aggregation ------------
// One thread per (edge, 4-channel group); HC = H*C, log2C selects the head.
__global__ void gat_edge_aggr(const float* __restrict__ h,
                              const float* __restrict__ a_src,
                              const float* __restrict__ a_dst,
                              const unsigned* __restrict__ mu,
                              const float* __restrict__ ssum,
                              float* __restrict__ acc,
                              const long long* __restrict__ srcp,
                              const long long* __restrict__ dstp,
                              int E, int Nn, int H, int HC, int log2C) {
    long long i = (long long)blockIdx.x * blockDim.x + threadIdx.x;
    int groups = HC >> 2;
    long long tot = (long long)(E + Nn) * groups;
    if (i >= tot) return;
    int e   = (int)(i / groups);
    int c4  = ((int)(i - (long long)e * groups)) << 2;
    int hh  = c4 >> log2C;
    int s, d; get_edge(srcp, dstp, e, E, s, d);

    float v = a_src[s * H + hh] + a_dst[d * H + hh];
    v = v > 0.f ? v : NEG_SLOPE * v;
    float w = __expf(v - dec_ord(mu[d * H + hh]));
    float alpha = w / (ssum[d * H + hh] + SM_EPS);

    const float4 hv = *reinterpret_cast<const float4*>(h + (size_t)s * HC + c4);
    float* ap = acc + (size_t)d * HC + c4;
    atomicAdd(ap + 0, hv.x * alpha);
    atomicAdd(ap + 1, hv.y * alpha);
    atomicAdd(ap + 2, hv.z * alpha);
    atomicAdd(ap + 3, hv.w * alpha);
}

// ---------------- epilogues --------------------------------------------------
__global__ void gat_bias_elu(float* __restrict__ acc, const float* __restrict__ bias,
                             int Nn, int HC) {
    int i = blockIdx.x * blockDim.x + threadIdx.x;
    if (i >= Nn * HC) return;
    float v = acc[i] + bias[i % HC];
    acc[i] = v > 0.f ? v : (__expf(v) - 1.0f);
}

__global__ void gat_bias_add(float* __restrict__ out, const float* __restrict__ bias,
                             int Nn, int C) {
    int i = blockIdx.x * blockDim.x + threadIdx.x;
    if (i >= Nn * C) return;
    out[i] += bias[i % C];
}

// ============================================================================
extern "C" void kernel_launch(void* const* d_in, const int* in_sizes, int n_in,
                              void* d_out, int out_size, void* d_ws, size_t ws_size,
                              hipStream_t stream) {
    const float*     x        = (const float*)d_in[0];
    const long long* ei       = (const long long*)d_in[1];   // [2,E] int64
    // d_in[2] = edge_weight: ignored by reference (edge_dim=None)
    const float*     W1       = (const float*)d_in[3];
    const float*     att_src1 = (const float*)d_in[4];
    const float*     att_dst1 = (const float*)d_in[5];
    const float*     bias1    = (const float*)d_in[6];
    const float*     W2       = (const float*)d_in[7];
    const float*     att_src2 = (const float*)d_in[8];
    const float*     att_dst2 = (const float*)d_in[9];
    const float*     bias2    = (const float*)d_in[10];
    float*           out      = (float*)d_out;

    const int Cin = 128, H1 = 8, C1 = 32, HC1 = 256, C2 = 64;
    const int Nn = in_sizes[0] / Cin;        // 50000
    const int E  = in_sizes[1] / 2;          // 800000
    const long long* srcp = ei;
    const long long* dstp = ei + E;

    // ---- workspace carve-out (all L2-resident: ~122 MB total) ----
    size_t off = 0;
    auto carve = [&](size_t bytes) -> void* {
        void* p = (char*)d_ws + off;
        off += (bytes + 255) & ~(size_t)255;
        return p;
    };
    float*    h1     = (float*)   carve((size_t)Nn * HC1 * 4);
    float*    as1    = (float*)   carve((size_t)Nn * H1 * 4);
    float*    ad1    = (float*)   carve((size_t)Nn * H1 * 4);
    unsigned* m1u    = (unsigned*)carve((size_t)Nn * H1 * 4);
    float*    s1     = (float*)   carve((size_t)Nn * H1 * 4);
    float*    acc1   = (float*)   carve((size_t)Nn * HC1 * 4);
    float*    h2     = (float*)   carve((size_t)Nn * C2 * 4);
    float*    as2    = (float*)   carve((size_t)Nn * 4);
    float*    ad2    = (float*)   carve((size_t)Nn * 4);
    unsigned* m2u    = (unsigned*)carve((size_t)Nn * 4);
    float*    s2     = (float*)   carve((size_t)Nn * 4);
    (void)ws_size;

    // ---- zero/init accumulators (memset is graph-capturable) ----
    hipMemsetAsync(m1u,  0, (size_t)Nn * H1 * 4, stream);  // enc_ord floor
    hipMemsetAsync(s1,   0, (size_t)Nn * H1 * 4, stream);
    hipMemsetAsync(acc1, 0, (size_t)Nn * HC1 * 4, stream);
    hipMemsetAsync(m2u,  0, (size_t)Nn * 4, stream);
    hipMemsetAsync(s2,   0, (size_t)Nn * 4, stream);
    hipMemsetAsync(out,  0, (size_t)Nn * C2 * 4, stream);

    const int TB = 256;
    auto cdiv = [](long long a, long long b) { return (int)((a + b - 1) / b); };
    const int Etot   = E + Nn;
    const int Mtiles = Nn >> 4;              // 3125 row tiles, 8 waves/block

    // ================= Layer 1 =================
    gat_gemm_wmma_lds<16><<<cdiv(Mtiles, 8), TB, 0, stream>>>(x, W1, h1, Nn, Cin);
    gat_att_scores<<<cdiv((long long)Nn * H1, TB), TB, 0, stream>>>(
        h1, att_src1, att_dst1, as1, ad1, Nn, H1, C1);
    gat_edge_max<<<cdiv((long long)Etot * H1, TB), TB, 0, stream>>>(
        as1, ad1, m1u, srcp, dstp, E, Nn, H1);
    gat_edge_sum<<<cdiv((long long)Etot * H1, TB), TB, 0, stream>>>(
        as1, ad1, m1u, s1, srcp, dstp, E, Nn, H1);
    gat_edge_aggr<<<cdiv((long long)Etot * (HC1 >> 2), TB), TB, 0, stream>>>(
        h1, as1, ad1, m1u, s1, acc1, srcp, dstp, E, Nn, H1, HC1, /*log2C=*/5);
    gat_bias_elu<<<cdiv((long long)Nn * HC1, TB), TB, 0, stream>>>(acc1, bias1, Nn, HC1);

    // ================= Layer 2 =================
    gat_gemm_wmma_lds<4><<<cdiv(Mtiles, 8), TB, 0, stream>>>(acc1, W2, h2, Nn, HC1);
    gat_att_scores<<<cdiv((long long)Nn, TB), TB, 0, stream>>>(
        h2, att_src2, att_dst2, as2, ad2, Nn, /*H=*/1, /*C=*/C2);
    gat_edge_max<<<cdiv((long long)Etot, TB), TB, 0, stream>>>(
        as2, ad2, m2u, srcp, dstp, E, Nn, /*H=*/1);
    gat_edge_sum<<<cdiv((long long)Etot, TB), TB, 0, stream>>>(
        as2, ad2, m2u, s2, srcp, dstp, E, Nn, /*H=*/1);
    gat_edge_aggr<<<cdiv((long long)Etot * (C2 >> 2), TB), TB, 0, stream>>>(
        h2, as2, ad2, m2u, s2, out, srcp, dstp, E, Nn, /*H=*/1, C2, /*log2C=*/6);
    gat_bias_add<<<cdiv((long long)Nn * C2, TB), TB, 0, stream>>>(out, bias2, Nn, C2);
}